// ThreeStageGNNLayer_33260226740761
// MI455X (gfx1250) — compile-verified
//
#include <hip/hip_runtime.h>

// CDNA5 / gfx1250 — wave32, WMMA f32 16x16x4 for full-precision fp32 GEMM.

typedef __attribute__((ext_vector_type(2))) float v2f;
typedef __attribute__((ext_vector_type(8))) float v8f;

#define DFEAT 128

// ---------------------------------------------------------------------------
// Zero a float region (graph-capture safe; d_ws is not pre-zeroed).
// ---------------------------------------------------------------------------
__global__ void zero_f4(float4* __restrict__ p, long n4) {
  long i = (long)blockIdx.x * blockDim.x + threadIdx.x;
  if (i < n4) p[i] = make_float4(0.f, 0.f, 0.f, 0.f);
}

// ---------------------------------------------------------------------------
// Edge scatter: one wave32 per edge; lane l handles floats [4l, 4l+3].
// accum[dst] += (ew ? ew[e] : 1) * hsrc[src];   deg[dst] += 1
// ---------------------------------------------------------------------------
__global__ __launch_bounds__(256)
void scatter_add(const float* __restrict__ hsrc,
                 const int*   __restrict__ src,
                 const int*   __restrict__ dst,
                 const float* __restrict__ ew,     // nullable
                 float* __restrict__ accum,
                 float* __restrict__ deg,
                 int E) {
  int gid  = blockIdx.x * blockDim.x + threadIdx.x;
  int edge = gid >> 5;
  int lane = gid & 31;
  if (edge >= E) return;

  int s = src[edge];
  int d = dst[edge];
  float w = ew ? ew[edge] : 1.0f;

  float4 v = *(const float4*)(hsrc + (size_t)s * DFEAT + lane * 4);
  float* ap = accum + (size_t)d * DFEAT + lane * 4;
  unsafeAtomicAdd(ap + 0, v.x * w);
  unsafeAtomicAdd(ap + 1, v.y * w);
  unsafeAtomicAdd(ap + 2, v.z * w);
  unsafeAtomicAdd(ap + 3, v.w * w);
  if (lane == 0) unsafeAtomicAdd(deg + d, 1.0f);
}

// ---------------------------------------------------------------------------
// Fused: out = relu(LN(hres + (accum/deg) @ W + bias))
// Block = 256 threads (8 waves), handles 16 rows x 128 cols.
// Each wave computes one 16x16 output tile via V_WMMA_F32_16X16X4_F32.
// ---------------------------------------------------------------------------
__global__ __launch_bounds__(256)
void gemm_ln_relu(const float* __restrict__ accum,
                  const float* __restrict__ deg,
                  const float* __restrict__ W,      // [128,128] row-major (k,n)
                  const float* __restrict__ bias,   // [128]
                  const float* __restrict__ hres,   // [N,128]
                  const float* __restrict__ ln_g,
                  const float* __restrict__ ln_b,
                  float* __restrict__ out,          // [N,128]
                  int N) {
  __shared__ float As[16][DFEAT + 4];
  __shared__ float Ds[16][DFEAT + 4];
  __shared__ float red[16][17];

  const int tid  = threadIdx.x;
  const int row0 = blockIdx.x * 16;
  if (row0 >= N) return;

  // ---- Stage A: load degree-normalized aggregate tile into LDS ----
  {
    int r  = tid >> 4;          // 0..15 row in tile
    int c  = (tid & 15) * 8;    // 8 consecutive floats per thread
    long gr = row0 + r;
    float s = 1.0f / fmaxf(deg[gr], 1.0f);
    const float4* src4 = (const float4*)(accum + gr * (long)DFEAT + c);
    float4 a0 = src4[0];
    float4 a1 = src4[1];
    As[r][c + 0] = a0.x * s;  As[r][c + 1] = a0.y * s;
    As[r][c + 2] = a0.z * s;  As[r][c + 3] = a0.w * s;
    As[r][c + 4] = a1.x * s;  As[r][c + 5] = a1.y * s;
    As[r][c + 6] = a1.z * s;  As[r][c + 7] = a1.w * s;
  }
  __syncthreads();

  // ---- Stage B: WMMA GEMM, one 16x16 tile per wave ----
  const int wave = tid >> 5;        // 0..7 -> column tile
  const int lane = tid & 31;
  const int n0   = wave * 16;
  const int m    = lane & 15;            // A row
  const int kh   = (lane >> 4) * 2;      // K sub-offset: lanes 0-15 -> K0/K1, 16-31 -> K2/K3
  const int nc   = lane & 15;            // B/D column

  v8f acc = {};
#pragma unroll
  for (int k0 = 0; k0 < DFEAT; k0 += 4) {
    v2f a, b;
    a.x = As[m][k0 + kh];
    a.y = As[m][k0 + kh + 1];
    const float* wp = W + (k0 + kh) * DFEAT + n0 + nc;
    b.x = wp[0];
    b.y = wp[DFEAT];
    acc = __builtin_amdgcn_wmma_f32_16x16x4_f32(
        /*neg_a=*/false, a, /*neg_b=*/false, b,
        /*c_mod=*/(short)0, acc, /*reuse_a=*/false, /*reuse_b=*/false);
  }

  // ---- Stage C: spill D tile to LDS (C/D layout: lanes<16 M=r, lanes>=16 M=r+8)
  {
    int mb = (lane >> 4) * 8;
#pragma unroll
    for (int r = 0; r < 8; ++r)
      Ds[mb + r][n0 + nc] = acc[r];
  }
  __syncthreads();

  // ---- Stage D: residual + bias + LayerNorm + ReLU ----
  {
    int r  = tid >> 4;
    int c0 = (tid & 15) * 8;
    long gr = row0 + r;
    float v[8];
    float psum = 0.f;
#pragma unroll
    for (int j = 0; j < 8; ++j) {
      float x = hres[gr * (long)DFEAT + c0 + j] + Ds[r][c0 + j] + bias[c0 + j];
      v[j] = x;
      psum += x;
    }
    red[r][tid & 15] = psum;
    __syncthreads();
    float mu = 0.f;
#pragma unroll
    for (int j = 0; j < 16; ++j) mu += red[r][j];
    mu *= (1.0f / DFEAT);
    __syncthreads();                       // all reads done before reuse
    float pvar = 0.f;
#pragma unroll
    for (int j = 0; j < 8; ++j) { float dd = v[j] - mu; pvar += dd * dd; }
    red[r][tid & 15] = pvar;
    __syncthreads();
    float var = 0.f;
#pragma unroll
    for (int j = 0; j < 16; ++j) var += red[r][j];
    var *= (1.0f / DFEAT);
    float rs = rsqrtf(var + 1e-5f);
#pragma unroll
    for (int j = 0; j < 8; ++j) {
      float y = (v[j] - mu) * rs * ln_g[c0 + j] + ln_b[c0 + j];
      out[gr * (long)DFEAT + c0 + j] = fmaxf(y, 0.f);
    }
  }
}

// ---------------------------------------------------------------------------
extern "C" void kernel_launch(void* const* d_in, const int* in_sizes, int n_in,
                              void* d_out, int out_size, void* d_ws, size_t ws_size,
                              hipStream_t stream) {
  const float* h_pin   = (const float*)d_in[0];
  const float* h_net   = (const float*)d_in[1];
  const float* ow      = (const float*)d_in[2];
  const int*   p2n_src = (const int*)d_in[3];
  const int*   p2n_dst = (const int*)d_in[4];
  const int*   n2n_src = (const int*)d_in[5];
  const int*   n2n_dst = (const int*)d_in[6];
  const float* W_p2n = (const float*)d_in[7];
  const float* b_p2n = (const float*)d_in[8];
  const float* W_n2n = (const float*)d_in[9];
  const float* b_n2n = (const float*)d_in[10];
  const float* W_n2p = (const float*)d_in[11];
  const float* b_n2p = (const float*)d_in[12];
  const float* ln1_g = (const float*)d_in[13];
  const float* ln1_b = (const float*)d_in[14];
  const float* ln2_g = (const float*)d_in[15];
  const float* ln2_b = (const float*)d_in[16];
  const float* ln3_g = (const float*)d_in[17];
  const float* ln3_b = (const float*)d_in[18];

  const int n_pin = in_sizes[0] / DFEAT;   // 400000
  const int n_net = in_sizes[1] / DFEAT;   // 100000
  const int e_p2n = in_sizes[3];           // 400000
  const int e_n2n = in_sizes[5];           // 800000

  float* out_pin = (float*)d_out;
  float* out_net = (float*)d_out + (size_t)n_pin * DFEAT;

  // Workspace: accum sized for the largest stage (n_pin rows), then degrees.
  float* accum = (float*)d_ws;
  float* degW  = accum + (size_t)n_pin * DFEAT;

  auto zero = [&](float* p, long n) {
    long n4 = n / 4;                     // all counts here are multiples of 4
    long grid = (n4 + 255) / 256;
    zero_f4<<<dim3((unsigned)grid), dim3(256), 0, stream>>>((float4*)p, n4);
  };
  auto scatter_grid = [](int E) { return (unsigned)(((long)E * 32 + 255) / 256); };

  // ---------------- Stage 1: pin -> net ----------------
  zero(accum, (long)n_net * DFEAT);
  zero(degW, n_net);
  scatter_add<<<scatter_grid(e_p2n), 256, 0, stream>>>(
      h_pin, p2n_src, p2n_dst, nullptr, accum, degW, e_p2n);
  gemm_ln_relu<<<(n_net + 15) / 16, 256, 0, stream>>>(
      accum, degW, W_p2n, b_p2n, h_net, ln1_g, ln1_b, out_net, n_net);

  // ---------------- Stage 2: net -> net (edge weights) ----------------
  zero(accum, (long)n_net * DFEAT);
  zero(degW, n_net);
  scatter_add<<<scatter_grid(e_n2n), 256, 0, stream>>>(
      out_net, n2n_src, n2n_dst, ow, accum, degW, e_n2n);
  gemm_ln_relu<<<(n_net + 15) / 16, 256, 0, stream>>>(
      accum, degW, W_n2n, b_n2n, out_net, ln2_g, ln2_b, out_net, n_net);

  // ---------------- Stage 3: net -> pin (reversed relation) ----------------
  zero(accum, (long)n_pin * DFEAT);
  zero(degW, n_pin);
  scatter_add<<<scatter_grid(e_p2n), 256, 0, stream>>>(
      out_net, p2n_dst, p2n_src, nullptr, accum, degW, e_p2n);
  gemm_ln_relu<<<(n_pin + 15) / 16, 256, 0, stream>>>(
      accum, degW, W_n2p, b_n2p, h_pin, ln3_g, ln3_b, out_pin, n_pin);
}